// MultiHeadEncoderAttention_26525718020547
// MI455X (gfx1250) — compile-verified
//
#include <hip/hip_runtime.h>

#define B_   8
#define H_   16
#define LQ_  1024
#define LK_  1024
#define DM_  1024
#define DK_  64

typedef __attribute__((ext_vector_type(2))) float v2f;
typedef __attribute__((ext_vector_type(8))) float v8f;

__device__ __forceinline__ v8f wmma4(v2f a, v2f b, v8f c) {
  // V_WMMA_F32_16X16X4_F32 : D(16x16 f32) = A(16x4 f32) x B(4x16 f32) + C
  return __builtin_amdgcn_wmma_f32_16x16x4_f32(false, a, false, b, (short)0, c,
                                               false, false);
}

__device__ __forceinline__ v8f v8zero() {
  v8f z;
#pragma unroll
  for (int i = 0; i < 8; ++i) z[i] = 0.0f;
  return z;
}

// Async global -> LDS (CDNA5): direct DMA into LDS, tracked by ASYNCcnt.
// LDS byte address = low 32 bits of the generic pointer (LDS aperture).
__device__ __forceinline__ void async_ld_b128(const float* lds_dst,
                                              const float* gsrc) {
  asm volatile("global_load_async_to_lds_b128 %0, %1, off"
               :: "v"((unsigned)(size_t)lds_dst), "v"(gsrc)
               : "memory");
}
__device__ __forceinline__ void wait_async0() {
  asm volatile("s_wait_asynccnt 0x0" ::: "memory");
}

// ---------------------------------------------------------------------------
// Projection: out[b,h,row,0:64] = X[b,row,:] @ W[h,:,:] + bias[h,:]
// grid = (B*(L/32), 2) x 256 threads (8 waves).
// Workgroup owns a 32-row M-tile + half the (head,n-tile) combos; each wave
// accumulates 4 combos x 2 M-subtiles (8 C-tiles), reusing each W fragment
// across both M-subtiles. A-chunks double-buffered in LDS via async loads.
// ---------------------------------------------------------------------------
__global__ __launch_bounds__(256) void proj_kernel(const float* __restrict__ X,
                                                   const float* __restrict__ W,
                                                   const float* __restrict__ bias,
                                                   float* __restrict__ out) {
  __shared__ float As[2][32][68];   // 272B row stride: 16B-aligned, bank-spread

  const int b     = blockIdx.x >> 5;      // 32 M-tiles per batch row-block
  const int mt    = blockIdx.x & 31;
  const int qbase = mt * 32;
  const int cgrp  = blockIdx.y;           // 0/1 : combos [0,32) or [32,64)
  const int lane  = threadIdx.x & 31;
  const int wave  = threadIdx.x >> 5;
  const int half  = lane >> 4;
  const int l16   = lane & 15;

  v8f acc[4][2];
#pragma unroll
  for (int j = 0; j < 4; ++j) {
    acc[j][0] = v8zero();
    acc[j][1] = v8zero();
  }

  auto issue = [&](int kb, int buf) {
#pragma unroll
    for (int r = 0; r < 2; ++r) {         // 32x64 floats = 512 float4
      const int fidx = threadIdx.x * 4 + r * 1024;
      const int row  = fidx >> 6;
      const int col  = fidx & 63;
      async_ld_b128(&As[buf][row][col],
                    X + (size_t)(b * LQ_ + qbase + row) * DM_ + kb * 64 + col);
    }
  };

  issue(0, 0);
  for (int kb = 0; kb < 16; ++kb) {       // K = 1024 in chunks of 64
    const int cur = kb & 1;
    wait_async0();                        // buffer `cur` landed (this thread)
    __syncthreads();                      // ... and for every thread
    if (kb < 15) issue(kb + 1, cur ^ 1);  // prefetch next chunk
    const int kbase = kb * 64;

#pragma unroll 4
    for (int kk = 0; kk < 16; ++kk) {     // 16 WMMA K-steps of 4
      const int c0 = 4 * kk + 2 * half;
      v2f a0, a1;                         // A-layout, two 16-row M-subtiles
      a0.x = As[cur][l16][c0];       a0.y = As[cur][l16][c0 + 1];
      a1.x = As[cur][16 + l16][c0];  a1.y = As[cur][16 + l16][c0 + 1];
      const int krow = kbase + c0;
#pragma unroll
      for (int j = 0; j < 4; ++j) {
        const int combo = cgrp * 32 + wave * 4 + j;
        const int h   = combo >> 2;
        const int nt  = combo & 3;
        const int col = nt * 16 + l16;
        const float* wp = W + (size_t)(h * DM_ + krow) * DK_ + col;
        v2f bb;                           // B fragment reused for both M-subtiles
        bb.x = wp[0];
        bb.y = wp[DK_];
        acc[j][0] = wmma4(a0, bb, acc[j][0]);
        acc[j][1] = wmma4(a1, bb, acc[j][1]);
      }
    }
  }

  // bias + store  (D-layout: M = i + 8*half, N = l16)
#pragma unroll
  for (int j = 0; j < 4; ++j) {
    const int combo = cgrp * 32 + wave * 4 + j;
    const int h   = combo >> 2;
    const int nt  = combo & 3;
    const int col = nt * 16 + l16;
    const float bvl = bias[h * DK_ + col];
#pragma unroll
    for (int mt2 = 0; mt2 < 2; ++mt2)
#pragma unroll
      for (int i = 0; i < 8; ++i) {
        const int q = qbase + mt2 * 16 + i + 8 * half;
        out[(size_t)((b * H_ + h) * LQ_ + q) * DK_ + col] = acc[j][mt2][i] + bvl;
      }
  }
}

// ---------------------------------------------------------------------------
// Flash attention: grid.x = B*H*(LQ/128), 256 threads (8 waves).
// Each wave owns 16 q-rows with online softmax; K/V double-buffered in LDS
// via async-to-LDS DMA.
// ---------------------------------------------------------------------------
__global__ __launch_bounds__(256) void attn_kernel(const float* __restrict__ Qb,
                                                   const float* __restrict__ Kb,
                                                   const float* __restrict__ Vb,
                                                   const int* __restrict__ mask,
                                                   float* __restrict__ out) {
  __shared__ float Ks[2][64][68];
  __shared__ float Vs[2][64][68];
  __shared__ float Ps[8][16][18];   // per-wave P shuffle scratch (D->A layout)

  const int t    = blockIdx.x;
  const int qblk = t & 7;
  const int h    = (t >> 3) & 15;
  const int b    = t >> 7;
  const int lane = threadIdx.x & 31;
  const int wave = threadIdx.x >> 5;
  const int half = lane >> 4;
  const int l16  = lane & 15;
  const int qbase = qblk * 128 + wave * 16;

  // This wave's Q tile (16 x 64) in A-layout fragments.
  const float* qptr = Qb + (size_t)((b * H_ + h) * LQ_ + qbase) * DK_;
  v2f aq[16];
#pragma unroll
  for (int kk = 0; kk < 16; ++kk) {
    const int c0 = 4 * kk + 2 * half;
    const float* p = qptr + (size_t)l16 * DK_ + c0;
    aq[kk].x = p[0];
    aq[kk].y = p[1];
  }

  float m[8], lsum[8];
#pragma unroll
  for (int i = 0; i < 8; ++i) { m[i] = -1e30f; lsum[i] = 0.0f; }
  v8f acc[4];
#pragma unroll
  for (int nt = 0; nt < 4; ++nt) acc[nt] = v8zero();

  const float scale = 0.125f;  // 1/sqrt(64)

  auto issue_kv = [&](int sb, int buf) {
#pragma unroll
    for (int r = 0; r < 4; ++r) {        // 64x64 floats = 1024 float4 each
      const int fidx = threadIdx.x * 4 + r * 1024;
      const int row  = fidx >> 6;
      const int col  = fidx & 63;
      const size_t g = (size_t)((b * H_ + h) * LK_ + sb * 64 + row) * DK_ + col;
      async_ld_b128(&Ks[buf][row][col], Kb + g);
      async_ld_b128(&Vs[buf][row][col], Vb + g);
    }
  };

  issue_kv(0, 0);
  for (int sb = 0; sb < 16; ++sb) {      // LK in blocks of 64
    const int cur = sb & 1;
    wait_async0();
    __syncthreads();
    if (sb < 15) issue_kv(sb + 1, cur ^ 1);
    const int sbase = sb * 64;

    for (int sc = 0; sc < 4; ++sc) {     // 16-wide s-chunks
      const int s0 = sc * 16;

      // S = Q K^T : B fragment is the K^T chunk read row-major from LDS
      v8f s = v8zero();
#pragma unroll 4
      for (int kk = 0; kk < 16; ++kk) {
        const int c0 = 4 * kk + 2 * half;
        v2f bk;
        bk.x = Ks[cur][s0 + l16][c0];
        bk.y = Ks[cur][s0 + l16][c0 + 1];
        s = wmma4(aq[kk], bk, s);
      }

      // scale + mask (mask applied after scaling, matching reference)
      const int scol = sbase + s0 + l16;
      float sv[8];
#pragma unroll
      for (int i = 0; i < 8; ++i) {
        const int q  = qbase + i + 8 * half;
        const int mv = mask[(size_t)(b * LQ_ + q) * LK_ + scol];
        sv[i] = (mv == 0) ? -1e30f : s[i] * scale;
      }

      // online softmax; D-layout row ownership matches the stats lanes
      float alpha[8], p[8];
#pragma unroll
      for (int i = 0; i < 8; ++i) {
        float cm = sv[i];
        for (int off = 1; off < 16; off <<= 1)
          cm = fmaxf(cm, __shfl_xor(cm, off));
        const float mnew = fmaxf(m[i], cm);
        alpha[i] = __expf(m[i] - mnew);
        p[i] = __expf(sv[i] - mnew);
        float rs = p[i];
        for (int off = 1; off < 16; off <<= 1)
          rs += __shfl_xor(rs, off);
        lsum[i] = lsum[i] * alpha[i] + rs;
        m[i] = mnew;
      }
#pragma unroll
      for (int nt = 0; nt < 4; ++nt)
#pragma unroll
        for (int i = 0; i < 8; ++i)
          acc[nt][i] *= alpha[i];

      // P: D-layout -> row-major LDS -> A-layout (per-wave scratch)
#pragma unroll
      for (int i = 0; i < 8; ++i)
        Ps[wave][i + 8 * half][l16] = p[i];
      asm volatile("s_wait_dscnt 0" ::: "memory");   // same-wave DS RAW

      // O += P * V
#pragma unroll
      for (int kk2 = 0; kk2 < 4; ++kk2) {
        const int c0 = 4 * kk2 + 2 * half;
        v2f ap;
        ap.x = Ps[wave][l16][c0];
        ap.y = Ps[wave][l16][c0 + 1];
        const int srow = s0 + c0;
#pragma unroll
        for (int nt = 0; nt < 4; ++nt) {
          const int col = nt * 16 + l16;
          v2f bv;
          bv.x = Vs[cur][srow][col];
          bv.y = Vs[cur][srow + 1][col];
          acc[nt] = wmma4(ap, bv, acc[nt]);
        }
      }
    }
  }

  // epilogue: normalize; fully-masked rows (m never left -1e30) -> 0
#pragma unroll
  for (int i = 0; i < 8; ++i) {
    const int q = qbase + i + 8 * half;
    const float inv = (m[i] <= -5e29f) ? 0.0f : 1.0f / lsum[i];
#pragma unroll
    for (int nt = 0; nt < 4; ++nt) {
      const int col = nt * 16 + l16;
      out[(size_t)(b * LQ_ + q) * (H_ * DK_) + h * DK_ + col] = acc[nt][i] * inv;
    }
  }
}

extern "C" void kernel_launch(void* const* d_in, const int* in_sizes, int n_in,
                              void* d_out, int out_size, void* d_ws, size_t ws_size,
                              hipStream_t stream) {
  const float* dec  = (const float*)d_in[0];
  const float* enc  = (const float*)d_in[1];
  const int*   mask = (const int*)d_in[2];
  const float* Wq   = (const float*)d_in[3];
  const float* bq   = (const float*)d_in[4];
  const float* Wk   = (const float*)d_in[5];
  const float* bk   = (const float*)d_in[6];
  const float* Wv   = (const float*)d_in[7];
  const float* bv   = (const float*)d_in[8];
  float* out = (float*)d_out;

  const size_t per = (size_t)B_ * H_ * LQ_ * DK_;   // 8 Mi floats each
  float* Qbuf = (float*)d_ws;
  float* Kbuf = Qbuf + per;
  float* Vbuf = Kbuf + per;

  dim3 blk(256);
  proj_kernel<<<dim3(B_ * 32, 2), blk, 0, stream>>>(dec, Wq, bq, Qbuf);
  proj_kernel<<<dim3(B_ * 32, 2), blk, 0, stream>>>(enc, Wk, bk, Kbuf);
  proj_kernel<<<dim3(B_ * 32, 2), blk, 0, stream>>>(enc, Wv, bv, Vbuf);
  attn_kernel<<<dim3(B_ * H_ * (LQ_ / 128)), blk, 0, stream>>>(Qbuf, Kbuf, Vbuf,
                                                               mask, out);
}